// MAMBA2D_60232621359572
// MI455X (gfx1250) — compile-verified
//
#include <hip/hip_runtime.h>

// ---------------------------------------------------------------------------
// MI455X / gfx1250 implementation of the bidirectional 2D Mamba block.
// GEMMs run on V_WMMA_F32_16X16X4_F32 (exact fp32 matrix path, wave32).
// B-tile staged pair-major in LDS so each B fragment is one ds_load_b64.
// ---------------------------------------------------------------------------

typedef __attribute__((ext_vector_type(2))) float v2f;
typedef __attribute__((ext_vector_type(8))) float v8f;

#define D_MODEL  256
#define D_INNER  512
#define D_STATE  16
#define DT_RANK  16
#define NTOK     8192      // B*H*W = 2*64*64
#define LSEQ     64
#define NSEQ     128       // sequences per view (B*64)
#define XDBL_N   48        // dt_rank + 2*d_state

// ---------------------------------------------------------------------------
// Generic fp32 WMMA GEMM:  Out[M,N] (=|+=)  A[M,K] @ W[N,K]^T  (+ bias)
// Block = 256 threads (8 waves, wm 0..3 x wn 0..1), tile 64x64, K chunk 32.
// A 16x4 fragment layout (ISA 7.12.2): lane l: m=l%16; half h=l/16 holds
// K={2h, 2h+1} in vgpr {0,1}.  B 4x16 mirrors it; C/D: row = v + 8h.
// BsP pair-major layout: BsP[k>>1][2*n + (k&1)]; row stride 160 floats so
// the two half-waves' 32-word b64 windows hit disjoint LDS bank sets.
// ---------------------------------------------------------------------------
template <bool ACC, bool BIAS>
__global__ __launch_bounds__(256) void mamba2d_wmma_gemm(
    const float* __restrict__ A,      // M x K row-major
    const float* __restrict__ W,      // N x K row-major
    const float* __restrict__ bias,   // N (or nullptr)
    float* __restrict__ Out,          // M x N
    int M, int N, int K)
{
    __shared__ float As[64][33];
    __shared__ float BsP[16][160];    // 16 K-pairs x (2*64 + 32 pad)

    const int tid  = threadIdx.x;
    const int lane = tid & 31;
    const int wave = tid >> 5;
    const int wm   = wave >> 1;        // 0..3 : 16-row subtile
    const int wn   = wave & 1;         // 0..1 : 32-col subtile
    const int m0   = blockIdx.y * 64;
    const int n0   = blockIdx.x * 64;
    const int ml   = lane & 15;
    const int hh   = lane >> 4;        // half-wave select

    v8f acc0 = {0.f,0.f,0.f,0.f,0.f,0.f,0.f,0.f};
    v8f acc1 = {0.f,0.f,0.f,0.f,0.f,0.f,0.f,0.f};

    const int ldr = tid >> 2;          // 0..63
    const int ldc = (tid & 3) * 8;     // 0,8,16,24

    for (int k0 = 0; k0 < K; k0 += 32) {
        // stage A tile (64 x 32)
        {
            const int gr = m0 + ldr;
            #pragma unroll
            for (int j = 0; j < 8; ++j)
                As[ldr][ldc + j] = (gr < M) ? A[(size_t)gr * K + k0 + ldc + j] : 0.f;
        }
        // stage W tile (32 x 64) into pair-major BsP[k>>1][2n + (k&1)]
        {
            const int gn = n0 + ldr;
            #pragma unroll
            for (int j = 0; j < 8; ++j) {
                const int k = ldc + j;
                BsP[k >> 1][2 * ldr + (k & 1)] =
                    (gn < N) ? W[(size_t)gn * K + k0 + k] : 0.f;
            }
        }
        // prefetch next K-chunk while this one is consumed
        if (k0 + 32 < K) {
            const int gr = m0 + ldr;
            const int gn = n0 + ldr;
            if (gr < M) __builtin_prefetch(&A[(size_t)gr * K + k0 + 32 + ldc], 0, 1);
            if (gn < N) __builtin_prefetch(&W[(size_t)gn * K + k0 + 32 + ldc], 0, 1);
        }
        __syncthreads();

        #pragma unroll
        for (int kk = 0; kk < 32; kk += 4) {
            v2f a;
            a.x = As[wm * 16 + ml][kk + 2 * hh];
            a.y = As[wm * 16 + ml][kk + 2 * hh + 1];
            const int kp = (kk >> 1) + hh;
            const v2f b0 = *(const v2f*)&BsP[kp][2 * (wn * 32 + ml)];
            const v2f b1 = *(const v2f*)&BsP[kp][2 * (wn * 32 + 16 + ml)];
            acc0 = __builtin_amdgcn_wmma_f32_16x16x4_f32(
                       false, a, false, b0, (short)0, acc0, false, false);
            acc1 = __builtin_amdgcn_wmma_f32_16x16x4_f32(
                       false, a, false, b1, (short)0, acc1, false, false);
        }
        __syncthreads();
    }

    #pragma unroll
    for (int v = 0; v < 8; ++v) {
        const int row  = m0 + wm * 16 + 8 * hh + v;
        const int col0 = n0 + wn * 32 + ml;
        const int col1 = col0 + 16;
        if (row < M) {
            float r0 = acc0[v], r1 = acc1[v];
            if (col0 < N) {
                if (BIAS) r0 += bias[col0];
                if (ACC)  Out[(size_t)row * N + col0] += r0;
                else      Out[(size_t)row * N + col0]  = r0;
            }
            if (col1 < N) {
                if (BIAS) r1 += bias[col1];
                if (ACC)  Out[(size_t)row * N + col1] += r1;
                else      Out[(size_t)row * N + col1]  = r1;
            }
        }
    }
}

// ---------------------------------------------------------------------------
// Depthwise causal conv1d (direction-aware, flips fused) + SiLU.
// dir=+1: y[q] = sum_j w[3-j]*x[q-j]   dir=-1: y[q] = sum_j w[3-j]*x[q+j]
// pos_div / pstride select row (1,1) or column (64,64) sequencing.
// ---------------------------------------------------------------------------
__global__ __launch_bounds__(256) void mamba2d_conv_silu(
    const float* __restrict__ xz,     // NTOK x 2*D_INNER (xi half = cols [0,512))
    const float* __restrict__ cw,     // D_INNER x 4
    const float* __restrict__ cb,     // D_INNER
    float* __restrict__ xi,           // NTOK x D_INNER
    int dir, int pos_div, int pstride)
{
    const int idx = blockIdx.x * 256 + threadIdx.x;   // NTOK*D_INNER
    const int d   = idx & (D_INNER - 1);
    const int tok = idx >> 9;
    const int pos = (tok / pos_div) & (LSEQ - 1);

    float acc = cb[d];
    #pragma unroll
    for (int j = 0; j < 4; ++j) {
        const int p = pos - dir * j;
        if (p >= 0 && p < LSEQ)
            acc += cw[d * 4 + (3 - j)] *
                   xz[(size_t)(tok - dir * j * pstride) * (2 * D_INNER) + d];
    }
    xi[idx] = acc / (1.f + __expf(-acc));             // SiLU
}

// ---------------------------------------------------------------------------
// Selective scan, fused: dt-projection (K=16 dot) + softplus + recurrence +
// D skip + SiLU(z) gating. One block = 256 channels of one sequence.
// x_dbl row (dt/B/C, 48 floats) staged through LDS per step.
// ---------------------------------------------------------------------------
__global__ __launch_bounds__(256) void mamba2d_scan(
    const float* __restrict__ xz,     // NTOK x 1024 (z at col 512+d)
    const float* __restrict__ xi,     // NTOK x 512
    const float* __restrict__ xdbl,   // NTOK x 48
    const float* __restrict__ dt_w,   // 512 x 16
    const float* __restrict__ dt_b,   // 512
    const float* __restrict__ A_log,  // 512 x 16
    const float* __restrict__ Dp,     // 512
    float* __restrict__ yg,           // NTOK x 512 (gated output)
    int dir, int is_v)
{
    __shared__ float row[XDBL_N];
    const int d = (blockIdx.y << 8) + threadIdx.x;    // channel 0..511
    const int n = blockIdx.x;                         // sequence 0..127
    const int seq_base = is_v ? ((n >> 6) * 4096 + (n & 63)) : (n * LSEQ);
    const int pstride  = is_v ? 64 : 1;

    float dtw[DT_RANK], Aa[D_STATE], h[D_STATE];
    #pragma unroll
    for (int s = 0; s < D_STATE; ++s) {
        dtw[s] = dt_w[d * DT_RANK + s];
        Aa[s]  = -__expf(A_log[d * D_STATE + s]);
        h[s]   = 0.f;
    }
    const float dtb = dt_b[d];
    const float Dd  = Dp[d];

    for (int i = 0; i < LSEQ; ++i) {
        const int q   = (dir > 0) ? i : (LSEQ - 1 - i);
        const int tok = seq_base + q * pstride;
        if (threadIdx.x < XDBL_N)
            row[threadIdx.x] = xdbl[(size_t)tok * XDBL_N + threadIdx.x];
        __syncthreads();

        float dtr = dtb;
        #pragma unroll
        for (int r = 0; r < DT_RANK; ++r) dtr += row[r] * dtw[r];
        const float dt = (dtr > 20.f) ? dtr : log1pf(__expf(dtr));  // softplus

        const float u  = xi[(size_t)tok * D_INNER + d];
        const float du = dt * u;
        float y = 0.f;
        #pragma unroll
        for (int s = 0; s < D_STATE; ++s) {
            h[s] = __expf(dt * Aa[s]) * h[s] + du * row[DT_RANK + s];
            y   += h[s] * row[DT_RANK + D_STATE + s];
        }
        y += Dd * u;
        const float z = xz[(size_t)tok * (2 * D_INNER) + D_INNER + d];
        y *= z / (1.f + __expf(-z));                  // * silu(z)
        yg[(size_t)tok * D_INNER + d] = y;
        __syncthreads();
    }
}

// ---------------------------------------------------------------------------
// Fold out_proj into the final fc: Wcomb[n,k] = sum_j out_w[j,k]*fc_w[n,off+j]
// so final out += ygate @ Wcomb^T (one WMMA GEMM per direction).
// ---------------------------------------------------------------------------
__global__ __launch_bounds__(256) void mamba2d_wcomb(
    const float* __restrict__ out_w,  // 256 x 512
    const float* __restrict__ fc_w,   // 256 x 1024
    float* __restrict__ wcomb,        // 256 x 512 (N x K)
    int fc_off)
{
    const int idx = blockIdx.x * 256 + threadIdx.x;   // 256*512
    const int k   = idx & (D_INNER - 1);
    const int nn  = idx >> 9;
    float acc = 0.f;
    #pragma unroll 8
    for (int j = 0; j < D_MODEL; ++j)
        acc += out_w[(size_t)j * D_INNER + k] * fc_w[(size_t)nn * 1024 + fc_off + j];
    wcomb[(size_t)nn * D_INNER + k] = acc;
}

// ---------------------------------------------------------------------------
extern "C" void kernel_launch(void* const* d_in, const int* in_sizes, int n_in,
                              void* d_out, int out_size, void* d_ws, size_t ws_size,
                              hipStream_t stream)
{
    const float* x        = (const float*)d_in[0];
    const float* mh_in_w  = (const float*)d_in[1];
    const float* mh_cw    = (const float*)d_in[2];
    const float* mh_cb    = (const float*)d_in[3];
    const float* mh_xp    = (const float*)d_in[4];
    const float* mh_dtw   = (const float*)d_in[5];
    const float* mh_dtb   = (const float*)d_in[6];
    const float* mh_Alog  = (const float*)d_in[7];
    const float* mh_D     = (const float*)d_in[8];
    const float* mh_ow    = (const float*)d_in[9];
    const float* mv_in_w  = (const float*)d_in[10];
    const float* mv_cw    = (const float*)d_in[11];
    const float* mv_cb    = (const float*)d_in[12];
    const float* mv_xp    = (const float*)d_in[13];
    const float* mv_dtw   = (const float*)d_in[14];
    const float* mv_dtb   = (const float*)d_in[15];
    const float* mv_Alog  = (const float*)d_in[16];
    const float* mv_D     = (const float*)d_in[17];
    const float* mv_ow    = (const float*)d_in[18];
    const float* fc_w     = (const float*)d_in[19];
    const float* fc_b     = (const float*)d_in[20];
    float*       out      = (float*)d_out;

    float* ws    = (float*)d_ws;
    float* xz    = ws;                                  // 8192*1024
    float* xi    = xz   + (size_t)NTOK * 1024;          // 8192*512
    float* xdbl  = xi   + (size_t)NTOK * D_INNER;       // 8192*48
    float* yg    = xdbl + (size_t)NTOK * XDBL_N;        // 8192*512
    float* wcomb = yg   + (size_t)NTOK * D_INNER;       // 256*512

    const dim3 blk(256);
    const dim3 g_inproj(1024 / 64, NTOK / 64);          // N tiles x M tiles
    const dim3 g_xdbl(1, NTOK / 64);
    const dim3 g_final(D_MODEL / 64, NTOK / 64);
    const dim3 g_scan(NSEQ, D_INNER / 256);

    auto run_dir = [&](const float* cw, const float* cb, const float* xp,
                       const float* dtw, const float* dtb, const float* Alog,
                       const float* Dp, const float* ow,
                       int dir, int is_v, int fc_off, bool first) {
        const int pos_div = is_v ? 64 : 1;
        const int pstride = is_v ? 64 : 1;
        mamba2d_conv_silu<<<(NTOK * D_INNER) / 256, blk, 0, stream>>>(
            xz, cw, cb, xi, dir, pos_div, pstride);
        mamba2d_wmma_gemm<false, false><<<g_xdbl, blk, 0, stream>>>(
            xi, xp, nullptr, xdbl, NTOK, XDBL_N, D_INNER);
        mamba2d_scan<<<g_scan, blk, 0, stream>>>(
            xz, xi, xdbl, dtw, dtb, Alog, Dp, yg, dir, is_v);
        mamba2d_wcomb<<<(D_MODEL * D_INNER) / 256, blk, 0, stream>>>(
            ow, fc_w, wcomb, fc_off);
        if (first)
            mamba2d_wmma_gemm<false, true><<<g_final, blk, 0, stream>>>(
                yg, wcomb, fc_b, out, NTOK, D_MODEL, D_INNER);
        else
            mamba2d_wmma_gemm<true, false><<<g_final, blk, 0, stream>>>(
                yg, wcomb, nullptr, out, NTOK, D_MODEL, D_INNER);
    };

    // Horizontal set: in_proj once, then forward + backward passes.
    mamba2d_wmma_gemm<false, false><<<g_inproj, blk, 0, stream>>>(
        x, mh_in_w, nullptr, xz, NTOK, 2 * D_INNER, D_MODEL);
    run_dir(mh_cw, mh_cb, mh_xp, mh_dtw, mh_dtb, mh_Alog, mh_D, mh_ow,
            +1, 0, 512, true);   // h1 -> fc cols [512,768)
    run_dir(mh_cw, mh_cb, mh_xp, mh_dtw, mh_dtb, mh_Alog, mh_D, mh_ow,
            -1, 0, 768, false);  // h2 -> fc cols [768,1024)

    // Vertical set: recompute xz with mv weights (reuse workspace).
    mamba2d_wmma_gemm<false, false><<<g_inproj, blk, 0, stream>>>(
        x, mv_in_w, nullptr, xz, NTOK, 2 * D_INNER, D_MODEL);
    run_dir(mv_cw, mv_cb, mv_xp, mv_dtw, mv_dtb, mv_Alog, mv_D, mv_ow,
            +1, 1, 0,   false);  // v1 -> fc cols [0,256)
    run_dir(mv_cw, mv_cb, mv_xp, mv_dtw, mv_dtb, mv_Alog, mv_D, mv_ow,
            -1, 1, 256, false);  // v2 -> fc cols [256,512)
}